// GATModule_81346680587017
// MI455X (gfx1250) — compile-verified
//
#include <hip/hip_runtime.h>
#include <hip/hip_bf16.h>

// ---------------------------------------------------------------------------
// GAT layer for MI455X (gfx1250, wave32).
//   xh = x @ W           : WMMA f32 16x16x4 GEMM, LDS-staged A, 2 tiles/wave
//   a_src/a_dst          : per-node dot products
//   scatter softmax      : uint-encoded atomicMax pass + f32 atomicAdd pass
//   aggregation          : 1 wave32 per edge, float4 gather + f32 atomic scatter
//   finalize             : += bias, ReLU
// ---------------------------------------------------------------------------

#define IN_C   128
#define OUTF   128   // HEADS*OUT_C
#define HEADS  4
#define OUT_C  32
#define NEG_SLOPE 0.2f

#define MT   32      // rows per block (two 16-row WMMA tiles per wave)
#define LDA  132     // padded LDS row stride (floats): banks = (l*4 + k) % 64

typedef float v2f __attribute__((ext_vector_type(2)));
typedef float v8f __attribute__((ext_vector_type(8)));

// ---- monotonic float<->uint encoding for atomic max on floats --------------
__device__ __forceinline__ unsigned enc_f32(float f) {
    unsigned u = __float_as_uint(f);
    return (u & 0x80000000u) ? ~u : (u | 0x80000000u);
}
__device__ __forceinline__ float dec_f32(unsigned k) {
    return __uint_as_float((k & 0x80000000u) ? (k ^ 0x80000000u) : ~k);
}
#define ENC_NEG_INF 0x007FFFFFu   // enc_f32(-inf)

__device__ __forceinline__ float leaky(float v) {
    return v > 0.0f ? v : NEG_SLOPE * v;
}

// ---------------------------------------------------------------------------
// Kernel 0: zero/neg-inf initialization of accumulators and output.
// ---------------------------------------------------------------------------
__global__ void gat_init(float* __restrict__ out, unsigned* __restrict__ m_enc,
                         float* __restrict__ s_sum, int n) {
    int i = blockIdx.x * blockDim.x + threadIdx.x;
    if (i < n * OUTF) out[i] = 0.0f;
    if (i < n * HEADS) { m_enc[i] = ENC_NEG_INF; s_sum[i] = 0.0f; }
}

// ---------------------------------------------------------------------------
// Kernel 1: xh = x @ W via V_WMMA_F32_16X16X4_F32.
// Block = 256 threads (8 waves) handles a 32-row strip of x.
// The strip is staged once in LDS (padded stride 132 -> conflict-free
// K-column reads) and shared by all 8 waves. Wave w computes the two 16x16
// output tiles at columns [16w,16w+16) for rows [0,16) and [16,32) of the
// strip, reusing one B fragment per k-step (2 WMMAs per B load).
//
// A frag (16x4 f32, 2 VGPRs): lane {half,l}; row = l; v2f = {K=k0+2h, K=k0+2h+1}
// B frag (4x16 f32, 2 VGPRs): col = l;  v2f = {W[k0+2h][col], W[k0+2h+1][col]}
// C/D  (16x16 f32, 8 VGPRs):  c[r] -> row r + 8*half, col = l
// ---------------------------------------------------------------------------
__global__ __launch_bounds__(256)
void gat_gemm_wmma(const float* __restrict__ x, const float* __restrict__ W,
                   float* __restrict__ xh, int n) {
    __shared__ float As[MT * LDA];   // 32*132*4 = 16.9 KB of the 320 KB LDS

    const int lane = threadIdx.x & 31;
    const int wave = threadIdx.x >> 5;      // 0..7 -> column tile
    const int half = lane >> 4;             // 0: lanes 0-15, 1: lanes 16-31
    const int l    = lane & 15;
    const int row0 = blockIdx.x * MT;
    const int colbase = wave * 16;

    // ---- stage A strip: 32 rows x 128 cols, float4 per thread x 4 ----------
    #pragma unroll
    for (int f = threadIdx.x; f < MT * (IN_C / 4); f += 256) {
        const int r  = f >> 5;              // 0..31
        const int c4 = (f & 31) * 4;        // 0,4,...,124
        int rr = row0 + r;
        if (rr >= n) rr = n - 1;            // clamp (N%32==0 so never taken)
        const float4 v = *(const float4*)(x + (size_t)rr * IN_C + c4);
        float* dst = As + r * LDA + c4;
        dst[0] = v.x; dst[1] = v.y; dst[2] = v.z; dst[3] = v.w;
    }
    __syncthreads();

    // ---- WMMA main loop ----------------------------------------------------
    v8f c0 = {}, c1 = {};
    #pragma unroll 4
    for (int k0 = 0; k0 < IN_C; k0 += 4) {
        const int ka = k0 + half * 2;
        v2f b;
        b.x = W[(size_t)(ka + 0) * OUTF + colbase + l];
        b.y = W[(size_t)(ka + 1) * OUTF + colbase + l];
        const v2f a0 = *(const v2f*)(As + l * LDA + ka);          // rows 0..15
        const v2f a1 = *(const v2f*)(As + (l + 16) * LDA + ka);   // rows 16..31
        // 8-arg form: (neg_a, A, neg_b, B, c_mod, C, reuse_a, reuse_b)
        c0 = __builtin_amdgcn_wmma_f32_16x16x4_f32(false, a0, false, b,
                                                   (short)0, c0, false, false);
        c1 = __builtin_amdgcn_wmma_f32_16x16x4_f32(false, a1, false, b,
                                                   (short)0, c1, false, false);
    }

    // ---- epilogue ----------------------------------------------------------
    float* __restrict__ o0 = xh + (size_t)(row0 + half * 8) * OUTF + colbase + l;
    if (row0 + MT <= n) {                   // uniform fast path (always, here)
        #pragma unroll
        for (int r = 0; r < 8; ++r) {
            o0[(size_t)r * OUTF]        = c0[r];
            o0[(size_t)(r + 16) * OUTF] = c1[r];
        }
    } else {
        #pragma unroll
        for (int r = 0; r < 8; ++r) {
            const int rowa = row0 + r + half * 8;
            if (rowa      < n) xh[(size_t)rowa * OUTF + colbase + l]        = c0[r];
            if (rowa + 16 < n) xh[(size_t)(rowa + 16) * OUTF + colbase + l] = c1[r];
        }
    }
}

// ---------------------------------------------------------------------------
// Kernel 2: per-node attention logits  a_src[n,h], a_dst[n,h].
// One thread per (node, head).
// ---------------------------------------------------------------------------
__global__ void gat_node_logits(const float* __restrict__ xh,
                                const float* __restrict__ att_src,
                                const float* __restrict__ att_dst,
                                float* __restrict__ a_src,
                                float* __restrict__ a_dst, int n) {
    int i = blockIdx.x * blockDim.x + threadIdx.x;
    if (i >= n * HEADS) return;
    const int h = i & (HEADS - 1);
    const float* __restrict__ v = xh + (size_t)(i >> 2) * OUTF + h * OUT_C;
    float ss = 0.0f, sd = 0.0f;
    #pragma unroll 8
    for (int cidx = 0; cidx < OUT_C; ++cidx) {
        const float f = v[cidx];
        ss = fmaf(f, att_src[h * OUT_C + cidx], ss);
        sd = fmaf(f, att_dst[h * OUT_C + cidx], sd);
    }
    a_src[i] = ss;
    a_dst[i] = sd;
}

// ---------------------------------------------------------------------------
// Kernel 3: scatter-softmax max pass. One thread per edge, all 4 heads
// via float4 loads (head buffers are 16B-aligned by construction).
// ---------------------------------------------------------------------------
__global__ void gat_edge_max(const long long* __restrict__ ei,
                             const float* __restrict__ a_src,
                             const float* __restrict__ a_dst,
                             unsigned* __restrict__ m_enc, int E, int n) {
    int e = blockIdx.x * blockDim.x + threadIdx.x;
    if (e >= E + n) return;
    long long s, d;
    if (e < E) { s = ei[e]; d = ei[(size_t)E + e]; } else { s = d = e - E; }
    const float4 as = *(const float4*)(a_src + s * HEADS);
    const float4 ad = *(const float4*)(a_dst + d * HEADS);
    unsigned* __restrict__ mp = m_enc + d * HEADS;
    atomicMax(mp + 0, enc_f32(leaky(as.x + ad.x)));
    atomicMax(mp + 1, enc_f32(leaky(as.y + ad.y)));
    atomicMax(mp + 2, enc_f32(leaky(as.z + ad.z)));
    atomicMax(mp + 3, enc_f32(leaky(as.w + ad.w)));
}

// ---------------------------------------------------------------------------
// Kernel 4: scatter-softmax sum pass. One thread per edge, 4 heads.
// ---------------------------------------------------------------------------
__global__ void gat_edge_sum(const long long* __restrict__ ei,
                             const float* __restrict__ a_src,
                             const float* __restrict__ a_dst,
                             const unsigned* __restrict__ m_enc,
                             float* __restrict__ s_sum, int E, int n) {
    int e = blockIdx.x * blockDim.x + threadIdx.x;
    if (e >= E + n) return;
    long long s, d;
    if (e < E) { s = ei[e]; d = ei[(size_t)E + e]; } else { s = d = e - E; }
    const float4 as = *(const float4*)(a_src + s * HEADS);
    const float4 ad = *(const float4*)(a_dst + d * HEADS);
    const unsigned* __restrict__ mp = m_enc + d * HEADS;
    float* __restrict__ sp = s_sum + d * HEADS;
    atomicAdd(sp + 0, __expf(leaky(as.x + ad.x) - dec_f32(mp[0])));
    atomicAdd(sp + 1, __expf(leaky(as.y + ad.y) - dec_f32(mp[1])));
    atomicAdd(sp + 2, __expf(leaky(as.z + ad.z) - dec_f32(mp[2])));
    atomicAdd(sp + 3, __expf(leaky(as.w + ad.w) - dec_f32(mp[3])));
}

// ---------------------------------------------------------------------------
// Kernel 5: weighted aggregation. One wave32 per edge; each lane handles 4
// consecutive channels (float4), head = lane>>3. Scatter via f32 atomics.
// Working set (xh 51MB + out 51MB + logits) is L2-resident (192 MB).
// ---------------------------------------------------------------------------
__global__ __launch_bounds__(256)
void gat_edge_aggregate(const long long* __restrict__ ei,
                        const float* __restrict__ a_src,
                        const float* __restrict__ a_dst,
                        const unsigned* __restrict__ m_enc,
                        const float* __restrict__ s_sum,
                        const float* __restrict__ xh,
                        float* __restrict__ out, int E, int n) {
    const int gw = (blockIdx.x * blockDim.x + threadIdx.x) >> 5;  // edge id
    if (gw >= E + n) return;
    const int lane = threadIdx.x & 31;
    long long s, d;
    if (gw < E) { s = ei[gw]; d = ei[(size_t)E + gw]; } else { s = d = gw - E; }

    const float* __restrict__ src_row = xh + (size_t)s * OUTF;
    __builtin_prefetch(src_row, 0, 0);                 // global_prefetch_b8

    const int h = lane >> 3;                            // 32 channels per head
    const float logit = leaky(a_src[s * HEADS + h] + a_dst[d * HEADS + h]);
    const float alpha = __expf(logit - dec_f32(m_enc[d * HEADS + h]))
                        / s_sum[d * HEADS + h];

    const float4 v = ((const float4*)src_row)[lane];
    float* __restrict__ o = out + (size_t)d * OUTF + lane * 4;
    atomicAdd(o + 0, alpha * v.x);
    atomicAdd(o + 1, alpha * v.y);
    atomicAdd(o + 2, alpha * v.z);
    atomicAdd(o + 3, alpha * v.w);
}

// ---------------------------------------------------------------------------
// Kernel 6: out = relu(out + bias)
// ---------------------------------------------------------------------------
__global__ void gat_finalize(float* __restrict__ out,
                             const float* __restrict__ bias, int n) {
    int i = blockIdx.x * blockDim.x + threadIdx.x;
    if (i >= n * OUTF) return;
    out[i] = fmaxf(out[i] + bias[i & (OUTF - 1)], 0.0f);
}

// ---------------------------------------------------------------------------
extern "C" void kernel_launch(void* const* d_in, const int* in_sizes, int n_in,
                              void* d_out, int out_size, void* d_ws, size_t ws_size,
                              hipStream_t stream) {
    const float*     x       = (const float*)d_in[0];
    const long long* ei      = (const long long*)d_in[1];  // int64 [2,E]
    const float*     W       = (const float*)d_in[2];
    const float*     att_src = (const float*)d_in[3];
    const float*     att_dst = (const float*)d_in[4];
    const float*     bias    = (const float*)d_in[5];
    float*           out     = (float*)d_out;

    const int n = in_sizes[0] / IN_C;
    const int E = in_sizes[1] / 2;

    // workspace layout (all 16B-aligned: sizes are multiples of 16 bytes)
    char* ws = (char*)d_ws;
    float*    xh    = (float*)ws;                 ws += (size_t)n * OUTF  * sizeof(float);
    float*    a_src = (float*)ws;                 ws += (size_t)n * HEADS * sizeof(float);
    float*    a_dst = (float*)ws;                 ws += (size_t)n * HEADS * sizeof(float);
    unsigned* m_enc = (unsigned*)ws;              ws += (size_t)n * HEADS * sizeof(unsigned);
    float*    s_sum = (float*)ws;                 ws += (size_t)n * HEADS * sizeof(float);

    const int T = 256;

    // 0: init out / max / sum
    gat_init<<<(n * OUTF + T - 1) / T, T, 0, stream>>>(out, m_enc, s_sum, n);

    // 1: WMMA GEMM  (32-row strip per block; N%32==0 here)
    gat_gemm_wmma<<<(n + MT - 1) / MT, T, 0, stream>>>(x, W, xh, n);

    // 2: node logits
    gat_node_logits<<<(n * HEADS + T - 1) / T, T, 0, stream>>>(
        xh, att_src, att_dst, a_src, a_dst, n);

    // 3/4: scatter softmax (one thread per edge incl. self-loops)
    const int nedges = E + n;
    gat_edge_max<<<(nedges + T - 1) / T, T, 0, stream>>>(ei, a_src, a_dst, m_enc, E, n);
    gat_edge_sum<<<(nedges + T - 1) / T, T, 0, stream>>>(ei, a_src, a_dst, m_enc, s_sum, E, n);

    // 5: aggregation (1 wave per edge -> 8 edges per 256-thread block)
    gat_edge_aggregate<<<(nedges + (T / 32) - 1) / (T / 32), T, 0, stream>>>(
        ei, a_src, a_dst, m_enc, s_sum, xh, out, E, n);

    // 6: bias + relu
    gat_finalize<<<(n * OUTF + T - 1) / T, T, 0, stream>>>(out, bias, n);
}